// GAT_29386166239467
// MI455X (gfx1250) — compile-verified
//
#include <hip/hip_runtime.h>

typedef __attribute__((ext_vector_type(2))) float v2f;
typedef __attribute__((ext_vector_type(8))) float v8f;

#define NEG_SLOPE 0.2f

// ---- order-preserving float <-> uint mapping for atomic max on floats ----
__device__ __forceinline__ unsigned f2ord(float f) {
    unsigned u = __float_as_uint(f);
    return (u & 0x80000000u) ? ~u : (u | 0x80000000u);
}
__device__ __forceinline__ float ord2f(unsigned u) {
    return (u & 0x80000000u) ? __uint_as_float(u & 0x7fffffffu)
                             : __uint_as_float(~u);
}
// f2ord(-inf) == 0x007FFFFF
#define ORD_NEG_INF 0x007FFFFFu

__device__ __forceinline__ float leaky(float v) {
    return v >= 0.f ? v : NEG_SLOPE * v;
}

// ------------------------- init -------------------------
__global__ void fill_f32(float* p, int n, float v) {
    int t = blockIdx.x * blockDim.x + threadIdx.x;
    if (t < n) p[t] = v;
}
__global__ void fill_u32(unsigned* p, int n, unsigned v) {
    int t = blockIdx.x * blockDim.x + threadIdx.x;
    if (t < n) p[t] = v;
}

// ------------------------- GEMM 1: h1 = x @ W1  ([n,128]x[128,128]) -------------------------
// One wave computes a 16x16 output tile with V_WMMA_F32_16X16X4_F32, K-loop of 32 steps.
// A frag (16x4 f32, 2 VGPR): lanes 0-15 -> M=lane, K={0,1}; lanes 16-31 -> M=lane-16, K={2,3}.
// B frag (4x16 f32, 2 VGPR): lanes 0-15 -> N=lane, K={0,1}; lanes 16-31 -> N=lane-16, K={2,3}.
// C/D (16x16 f32, 8 VGPR): vgpr r -> M = r + (lane>=16 ? 8 : 0), N = lane&15.
__global__ void gemm_x_w1(const float* __restrict__ x, const float* __restrict__ W,
                          float* __restrict__ h, int n, int mtiles) {
    int wave = (blockIdx.x * blockDim.x + threadIdx.x) >> 5;
    int lane = threadIdx.x & 31;
    int mt = wave >> 3;          // 8 column tiles (128 cols)
    int nt = wave & 7;
    if (mt >= mtiles) return;    // wave-uniform: EXEC stays all-1 for WMMA
    int l15 = lane & 15;
    int kh  = (lane >> 4) << 1;  // 0 or 2
    int row = mt * 16 + l15;
    int rrow = row < n ? row : n - 1;          // clamp tail reads
    const float* xa = x + (size_t)rrow * 128;
    const float* wb = W + nt * 16 + l15;
    v8f acc = {};
    #pragma unroll 4
    for (int k = 0; k < 128; k += 4) {
        v2f a, b;
        a.x = xa[k + kh];
        a.y = xa[k + kh + 1];
        b.x = wb[(size_t)(k + kh) * 128];
        b.y = wb[(size_t)(k + kh + 1) * 128];
        acc = __builtin_amdgcn_wmma_f32_16x16x4_f32(false, a, false, b,
                                                    (short)0, acc, false, false);
    }
    int mbase = mt * 16 + ((lane >> 4) << 3);
    float* out = h + (size_t)mbase * 128 + nt * 16 + l15;
    if (mt * 16 + 16 <= n) {     // wave-uniform full-tile fast path: no exec juggling
        #pragma unroll
        for (int r = 0; r < 8; r++) out[(size_t)r * 128] = acc[r];
    } else {
        #pragma unroll
        for (int r = 0; r < 8; r++)
            if (mbase + r < n) out[(size_t)r * 128] = acc[r];
    }
}

// ------------------------- per-node attention scalars, layer 1 -------------------------
__global__ void att1(const float* __restrict__ h1, const float* __restrict__ as,
                     const float* __restrict__ ad, float* __restrict__ a_s,
                     float* __restrict__ a_d, int n) {
    int t = blockIdx.x * blockDim.x + threadIdx.x;
    if (t >= n * 8) return;
    int node = t >> 3, hd = t & 7;
    const float* hp = h1 + (size_t)node * 128 + hd * 16;
    const float* sp = as + hd * 16;
    const float* dp = ad + hd * 16;
    float vs = 0.f, vd = 0.f;
    #pragma unroll
    for (int c = 0; c < 16; c++) { float v = hp[c]; vs += v * sp[c]; vd += v * dp[c]; }
    a_s[t] = vs; a_d[t] = vd;
}

// ------------------------- edge passes, layer 1 (8 heads) -------------------------
__device__ __forceinline__ void edge_sd(const int* ei, int e, int E, int& src, int& dst) {
    if (e < E) { src = ei[e]; dst = ei[E + e]; }
    else       { src = e - E; dst = e - E; }      // self loops
}

__global__ void edge_max1(const int* __restrict__ ei, const float* __restrict__ a_s,
                          const float* __restrict__ a_d, unsigned* __restrict__ m,
                          int E, int n) {
    int t = blockIdx.x * blockDim.x + threadIdx.x;
    if (t >= (E + n) * 8) return;
    int e = t >> 3, hd = t & 7, src, dst;
    edge_sd(ei, e, E, src, dst);
    float v = leaky(a_s[src * 8 + hd] + a_d[dst * 8 + hd]);
    atomicMax(&m[dst * 8 + hd], f2ord(v));
}

__global__ void edge_sum1(const int* __restrict__ ei, const float* __restrict__ a_s,
                          const float* __restrict__ a_d, const unsigned* __restrict__ m,
                          float* __restrict__ denom, int E, int n) {
    int t = blockIdx.x * blockDim.x + threadIdx.x;
    if (t >= (E + n) * 8) return;
    int e = t >> 3, hd = t & 7, src, dst;
    edge_sd(ei, e, E, src, dst);
    float v = leaky(a_s[src * 8 + hd] + a_d[dst * 8 + hd]);
    float ee = __expf(v - ord2f(m[dst * 8 + hd]));
    atomicAdd(&denom[dst * 8 + hd], ee);
}

__global__ void edge_agg1(const int* __restrict__ ei, const float* __restrict__ a_s,
                          const float* __restrict__ a_d, const unsigned* __restrict__ m,
                          const float* __restrict__ denom, const float* __restrict__ h1,
                          float* __restrict__ out, int E, int n) {
    int t = blockIdx.x * blockDim.x + threadIdx.x;
    if (t >= (E + n) * 8) return;
    int e = t >> 3, hd = t & 7, src, dst;
    edge_sd(ei, e, E, src, dst);
    float v = leaky(a_s[src * 8 + hd] + a_d[dst * 8 + hd]);
    float alpha = __expf(v - ord2f(m[dst * 8 + hd])) / denom[dst * 8 + hd];
    const float* hp = h1 + (size_t)src * 128 + hd * 16;
    float* op = out + (size_t)dst * 128 + hd * 16;
    #pragma unroll
    for (int c = 0; c < 16; c++) atomicAdd(&op[c], alpha * hp[c]);
}

// ------------------------- h2 = relu(out1 + b1), in place -------------------------
__global__ void relu_bias1(float* __restrict__ out1, const float* __restrict__ b1, int n) {
    int t = blockIdx.x * blockDim.x + threadIdx.x;
    if (t >= n * 128) return;
    float v = out1[t] + b1[t & 127];
    out1[t] = v > 0.f ? v : 0.f;
}

// ------------------------- GEMM 2: z = h2 @ W2  ([n,128]x[128,40], padded to 48) -------------------------
__global__ void gemm_h_w2(const float* __restrict__ h, const float* __restrict__ W,
                          float* __restrict__ z, int n, int mtiles) {
    int wave = (blockIdx.x * blockDim.x + threadIdx.x) >> 5;
    int lane = threadIdx.x & 31;
    int mt = wave / 3;           // 3 column tiles cover 40 (pad to 48)
    int nt = wave % 3;
    if (mt >= mtiles) return;
    int l15 = lane & 15;
    int kh  = (lane >> 4) << 1;
    int col = nt * 16 + l15;
    int row = mt * 16 + l15;
    int rrow = row < n ? row : n - 1;
    const float* ha = h + (size_t)rrow * 128;
    v8f acc = {};
    #pragma unroll 4
    for (int k = 0; k < 128; k += 4) {
        v2f a, b;
        a.x = ha[k + kh];
        a.y = ha[k + kh + 1];
        b.x = (col < 40) ? W[(size_t)(k + kh) * 40 + col] : 0.f;
        b.y = (col < 40) ? W[(size_t)(k + kh + 1) * 40 + col] : 0.f;
        acc = __builtin_amdgcn_wmma_f32_16x16x4_f32(false, a, false, b,
                                                    (short)0, acc, false, false);
    }
    if (col < 40) {
        int mbase = mt * 16 + ((lane >> 4) << 3);
        float* zp = z + (size_t)mbase * 40 + col;
        if (mt * 16 + 16 <= n) { // uniform full-tile fast path
            #pragma unroll
            for (int r = 0; r < 8; r++) zp[(size_t)r * 40] = acc[r];
        } else {
            #pragma unroll
            for (int r = 0; r < 8; r++)
                if (mbase + r < n) zp[(size_t)r * 40] = acc[r];
        }
    }
}

// ------------------------- per-node attention scalars, layer 2 (H=1, C=40) -------------------------
__global__ void att2(const float* __restrict__ z, const float* __restrict__ as,
                     const float* __restrict__ ad, float* __restrict__ a_s,
                     float* __restrict__ a_d, int n) {
    int t = blockIdx.x * blockDim.x + threadIdx.x;
    if (t >= n) return;
    const float* zp = z + (size_t)t * 40;
    float vs = 0.f, vd = 0.f;
    #pragma unroll
    for (int c = 0; c < 40; c++) { float v = zp[c]; vs += v * as[c]; vd += v * ad[c]; }
    a_s[t] = vs; a_d[t] = vd;
}

// ------------------------- edge passes, layer 2 -------------------------
__global__ void edge_max2(const int* __restrict__ ei, const float* __restrict__ a_s,
                          const float* __restrict__ a_d, unsigned* __restrict__ m,
                          int E, int n) {
    int t = blockIdx.x * blockDim.x + threadIdx.x;
    if (t >= E + n) return;
    int src, dst; edge_sd(ei, t, E, src, dst);
    atomicMax(&m[dst], f2ord(leaky(a_s[src] + a_d[dst])));
}

__global__ void edge_sum2(const int* __restrict__ ei, const float* __restrict__ a_s,
                          const float* __restrict__ a_d, const unsigned* __restrict__ m,
                          float* __restrict__ denom, int E, int n) {
    int t = blockIdx.x * blockDim.x + threadIdx.x;
    if (t >= E + n) return;
    int src, dst; edge_sd(ei, t, E, src, dst);
    float v = leaky(a_s[src] + a_d[dst]);
    atomicAdd(&denom[dst], __expf(v - ord2f(m[dst])));
}

__global__ void edge_agg2(const int* __restrict__ ei, const float* __restrict__ a_s,
                          const float* __restrict__ a_d, const unsigned* __restrict__ m,
                          const float* __restrict__ denom, const float* __restrict__ z,
                          float* __restrict__ out, int E, int n) {
    int t = blockIdx.x * blockDim.x + threadIdx.x;
    if (t >= (E + n) * 8) return;
    int e = t >> 3, sub = t & 7, src, dst;
    edge_sd(ei, e, E, src, dst);
    float v = leaky(a_s[src] + a_d[dst]);
    float alpha = __expf(v - ord2f(m[dst])) / denom[dst];
    int c0 = sub * 5;                           // 8 lanes x 5 channels = 40
    const float* zp = z + (size_t)src * 40 + c0;
    float* op = out + (size_t)dst * 40 + c0;
    #pragma unroll
    for (int c = 0; c < 5; c++) atomicAdd(&op[c], alpha * zp[c]);
}

// ------------------------- final: log_softmax(out2 + b2) -------------------------
__global__ void final_logsoftmax(const float* __restrict__ out2, const float* __restrict__ b2,
                                 float* __restrict__ y, int n) {
    int i = blockIdx.x * blockDim.x + threadIdx.x;
    if (i >= n) return;
    float v[40];
    float mx = -3.402823466e38f;
    #pragma unroll
    for (int c = 0; c < 40; c++) {
        v[c] = out2[(size_t)i * 40 + c] + b2[c];
        mx = v[c] > mx ? v[c] : mx;
    }
    float s = 0.f;
    #pragma unroll
    for (int c = 0; c < 40; c++) s += __expf(v[c] - mx);
    float lse = mx + __logf(s);
    #pragma unroll
    for (int c = 0; c < 40; c++) y[(size_t)i * 40 + c] = v[c] - lse;
}

// ------------------------- launcher -------------------------
extern "C" void kernel_launch(void* const* d_in, const int* in_sizes, int n_in,
                              void* d_out, int out_size, void* d_ws, size_t ws_size,
                              hipStream_t stream) {
    const float* x    = (const float*)d_in[0];
    const int*   ei   = (const int*)d_in[1];
    const float* W1   = (const float*)d_in[2];
    const float* as1  = (const float*)d_in[3];
    const float* ad1  = (const float*)d_in[4];
    const float* b1   = (const float*)d_in[5];
    const float* W2   = (const float*)d_in[6];
    const float* as2  = (const float*)d_in[7];
    const float* ad2  = (const float*)d_in[8];
    const float* b2   = (const float*)d_in[9];
    float* y = (float*)d_out;

    const int n = in_sizes[0] / 128;
    const int E = in_sizes[1] / 2;

    // workspace layout
    float*    h1     = (float*)d_ws;                       // n*128
    float*    a_s1   = h1 + (size_t)n * 128;               // n*8
    float*    a_d1   = a_s1 + (size_t)n * 8;               // n*8
    unsigned* m1     = (unsigned*)(a_d1 + (size_t)n * 8);  // n*8
    float*    denom1 = (float*)(m1 + (size_t)n * 8);       // n*8
    float*    out1   = denom1 + (size_t)n * 8;             // n*128 (becomes h2 after relu)
    float*    z2     = out1 + (size_t)n * 128;             // n*40
    float*    a_s2   = z2 + (size_t)n * 40;                // n
    float*    a_d2   = a_s2 + n;                           // n
    unsigned* m2     = (unsigned*)(a_d2 + n);              // n
    float*    denom2 = (float*)(m2 + n);                   // n
    float*    out2   = denom2 + n;                         // n*40

    const int B = 256;
    auto blk = [](int t, int b) { return (t + b - 1) / b; };

    // init accumulators
    fill_f32<<<blk(n * 128, B), B, 0, stream>>>(out1, n * 128, 0.f);
    fill_f32<<<blk(n * 8, B),   B, 0, stream>>>(denom1, n * 8, 0.f);
    fill_u32<<<blk(n * 8, B),   B, 0, stream>>>(m1, n * 8, ORD_NEG_INF);
    fill_f32<<<blk(n * 40, B),  B, 0, stream>>>(out2, n * 40, 0.f);
    fill_f32<<<blk(n, B),       B, 0, stream>>>(denom2, n, 0.f);
    fill_u32<<<blk(n, B),       B, 0, stream>>>(m2, n, ORD_NEG_INF);

    // layer 1
    int mtiles = (n + 15) / 16;
    gemm_x_w1<<<blk(mtiles * 8 * 32, B), B, 0, stream>>>(x, W1, h1, n, mtiles);
    att1<<<blk(n * 8, B), B, 0, stream>>>(h1, as1, ad1, a_s1, a_d1, n);

    int EA8 = (E + n) * 8;
    edge_max1<<<blk(EA8, B), B, 0, stream>>>(ei, a_s1, a_d1, m1, E, n);
    edge_sum1<<<blk(EA8, B), B, 0, stream>>>(ei, a_s1, a_d1, m1, denom1, E, n);
    edge_agg1<<<blk(EA8, B), B, 0, stream>>>(ei, a_s1, a_d1, m1, denom1, h1, out1, E, n);
    relu_bias1<<<blk(n * 128, B), B, 0, stream>>>(out1, b1, n);

    // layer 2
    gemm_h_w2<<<blk(mtiles * 3 * 32, B), B, 0, stream>>>(out1, W2, z2, n, mtiles);
    att2<<<blk(n, B), B, 0, stream>>>(z2, as2, ad2, a_s2, a_d2, n);

    int EA = E + n;
    edge_max2<<<blk(EA, B), B, 0, stream>>>(ei, a_s2, a_d2, m2, E, n);
    edge_sum2<<<blk(EA, B), B, 0, stream>>>(ei, a_s2, a_d2, m2, denom2, E, n);
    edge_agg2<<<blk(EA8, B), B, 0, stream>>>(ei, a_s2, a_d2, m2, denom2, z2, out2, E, n);

    final_logsoftmax<<<blk(n, B), B, 0, stream>>>(out2, b2, y, n);
}